// AudioPreprocessor_28080496181681
// MI455X (gfx1250) — compile-verified
//
#include <hip/hip_runtime.h>
#include <math.h>

#ifndef M_PI
#define M_PI 3.14159265358979323846
#endif

typedef __attribute__((ext_vector_type(2))) float v2f;
typedef __attribute__((ext_vector_type(8))) float v8f;

#define NTAPS 256                       // FIR truncation (pole radius ~0.925)
#define KCH   68                        // 272 window cols / 4-wide f32 WMMA k
#define RUN_TILES 32                    // 512 outputs per wave-run (divides 60000)
#define SLOTS (NTAPS + RUN_TILES * 16)  // 768 linear window slots
#define SSTRIDE 772                     // 768 + 4: conflict-free banks, 16B rows

// One-thread init kernel: impulse response of the biquad, double accumulation
// of the float32 coefficients (matches reference's float32 coefficient cast).
__global__ void build_impulse(float* hg, float b0, float b1, float b2,
                              float a1, float a2)
{
    if (threadIdx.x == 0 && blockIdx.x == 0) {
        double h1 = 0.0, h2 = 0.0;
        for (int n = 0; n < NTAPS; ++n) {
            double bn = (n == 0) ? (double)b0 : (n == 1) ? (double)b1
                      : (n == 2) ? (double)b2 : 0.0;
            double hn = bn - (double)a1 * h1 - (double)a2 * h2;
            hg[n] = (float)hn;
            h2 = h1; h1 = hn;
        }
    }
}

__global__ __launch_bounds__(32)
void biquad_wmma_fir(const float* __restrict__ x, float* __restrict__ y,
                     const float* __restrict__ hg, int T, int nbt, float gain)
{
    __shared__ float hsh[NTAPS];
    __shared__ float win[SSTRIDE * 16];

    const int lane = threadIdx.x & 31;
    const int half = lane >> 4;       // K sub-row selector within fragments
    const int col  = lane & 15;       // A: row m | B/D: batch column n

    // Stage h[0..255] into LDS (single wave per block; dscnt ordering suffices).
    for (int i = lane; i < NTAPS; i += 32) hsh[i] = hg[i];

    const int run      = blockIdx.x;
    const int b_base   = (run % nbt) * 16;
    const int runStart = (run / nbt) * (RUN_TILES * 16);

    // Tile-invariant Toeplitz A fragments.
    // Window w[j] = x[t0 - 256 + j]; A[m][j] = h[m + 256 - j] when 0<=d<256.
    // 16x4 f32 A layout: lanes 0-15 hold K = o,o+1; lanes 16-31 K = o+2,o+3.
    v2f aF[KCH];
#pragma unroll
    for (int kk = 0; kk < KCH; ++kk) {
        int o  = 4 * kk + 2 * half;
        int d0 = col + NTAPS - o;
        int d1 = d0 - 1;
        float A0 = (d0 >= 0 && d0 < NTAPS) ? hsh[d0] : 0.0f;
        float A1 = (d1 >= 0 && d1 < NTAPS) ? hsh[d1] : 0.0f;
        v2f t; t[0] = A0; t[1] = A1;
        aF[kk] = t;
    }

    // Preload history slots s in [0, 256) <-> t = runStart - 256 + s.
    // runStart == 0 (4 blocks total): zero history == IIR zero initial state.
    // Otherwise: unguarded, fully vectorized b128 path.
    if (runStart == 0) {
        for (int i = lane; i < NTAPS * 16; i += 32)
            win[(i >> 8) * SSTRIDE + (i & (NTAPS - 1))] = 0.0f;
    } else {
#pragma unroll
        for (int n = 0; n < 16; ++n) {
            const float* src = x + (size_t)(b_base + n) * T + runStart - NTAPS;
            float* dst = win + n * SSTRIDE;
            for (int s4 = lane * 4; s4 < NTAPS; s4 += 128) {
                float4 v = *(const float4*)(src + s4);
                *(float4*)(dst + s4) = v;
            }
        }
    }

    const float* xrow  = x + (size_t)(b_base + col) * T;
    float*       yrow  = y + (size_t)(b_base + col) * T;
    const float* bptr0 = win + col * SSTRIDE + 2 * half;          // B-frag base
    float*       sptr0 = win + col * SSTRIDE + NTAPS + 8 * half;  // stream base

    // Stream tile 0 (slots 256..271 <-> t = runStart..runStart+15).
    int t0 = runStart;
    {
        float4 pa = *(const float4*)(xrow + t0 + 8 * half);
        float4 pb = *(const float4*)(xrow + t0 + 8 * half + 4);
        *(float4*)(sptr0)     = pa;
        *(float4*)(sptr0 + 4) = pb;
    }

    for (int tt = 0; tt < RUN_TILES; ++tt) {
        const float* bptr = bptr0 + 16 * tt;   // slot 16*tt; +16*kk bytes -> imm

        // Prefetch next tile's samples into registers (uniform branch).
        float4 na, nb;
        if (tt + 1 < RUN_TILES) {
            na = *(const float4*)(xrow + t0 + 16 + 8 * half);
            nb = *(const float4*)(xrow + t0 + 16 + 8 * half + 4);
        }

        // Y(16x16) = A(16x272) * W(272x16): 68 f32 WMMAs on two interleaved
        // accumulator chains, fed by merged ds_load_2addr_b64 pairs.
        v8f acc0 = {};
        v8f acc1 = {};
#pragma unroll
        for (int kk = 0; kk < KCH; kk += 2) {
            v2f bF0 = *(const v2f*)(bptr + 4 * kk);
            v2f bF1 = *(const v2f*)(bptr + 4 * kk + 4);
            acc0 = __builtin_amdgcn_wmma_f32_16x16x4_f32(
                false, aF[kk],     false, bF0, (short)0, acc0, false, false);
            acc1 = __builtin_amdgcn_wmma_f32_16x16x4_f32(
                false, aF[kk + 1], false, bF1, (short)0, acc1, false, false);
        }
        v8f acc = acc0 + acc1;

        // Commit next tile to LDS (slots disjoint from this tile's reads).
        if (tt + 1 < RUN_TILES) {
            float* sp = sptr0 + 16 * (tt + 1);
            *(float4*)(sp)     = na;
            *(float4*)(sp + 4) = nb;
        }

        // D layout: VGPR r holds M = r + 8*half, N = col. Gain+clamp, store.
        {
            float4 q0, q1;
            q0.x = fminf(1.0f, fmaxf(-1.0f, acc[0] * gain));
            q0.y = fminf(1.0f, fmaxf(-1.0f, acc[1] * gain));
            q0.z = fminf(1.0f, fmaxf(-1.0f, acc[2] * gain));
            q0.w = fminf(1.0f, fmaxf(-1.0f, acc[3] * gain));
            q1.x = fminf(1.0f, fmaxf(-1.0f, acc[4] * gain));
            q1.y = fminf(1.0f, fmaxf(-1.0f, acc[5] * gain));
            q1.z = fminf(1.0f, fmaxf(-1.0f, acc[6] * gain));
            q1.w = fminf(1.0f, fmaxf(-1.0f, acc[7] * gain));
            float* dst = yrow + t0 + 8 * half;
            *(float4*)(dst)     = q0;
            *(float4*)(dst + 4) = q1;
        }

        t0 += 16;
    }
}

extern "C" void kernel_launch(void* const* d_in, const int* in_sizes, int n_in,
                              void* d_out, int out_size, void* d_ws, size_t ws_size,
                              hipStream_t stream)
{
    const float* x = (const float*)d_in[0];
    float* y = (float*)d_out;
    float* hg = (float*)d_ws;        // 256 floats of scratch for h[]

    const int T = 960000;
    const int B = in_sizes[0] / T;   // 64

    // torchaudio equalizer_biquad coefficients (sr = 48000 fixed by reference),
    // cast to float32 to match np.float32(c / a0).
    const double sr = 48000.0, fc = 1000.0, gdb = 3.0, Q = 0.707;
    const double w0 = 2.0 * M_PI * fc / sr;
    const double A  = pow(10.0, gdb / 40.0);
    const double al = sin(w0) / (2.0 * Q);
    const double a0 = 1.0 + al / A;
    const float b0 = (float)((1.0 + al * A) / a0);
    const float b1 = (float)((-2.0 * cos(w0)) / a0);
    const float b2 = (float)((1.0 - al * A) / a0);
    const float a1 = (float)((-2.0 * cos(w0)) / a0);
    const float a2 = (float)((1.0 - al / A) / a0);

    build_impulse<<<1, 32, 0, stream>>>(hg, b0, b1, b2, a1, a2);

    const int timeTiles = T / 16;                         // 60000
    const int nbt = B / 16;                               // 4
    const int totalRuns = (timeTiles / RUN_TILES) * nbt;  // 1875 * 4 = 7500

    biquad_wmma_fir<<<totalRuns, 32, 0, stream>>>(x, y, hg, T, nbt, 0.8f);
}